// NeighborAttention_53523882443206
// MI455X (gfx1250) — compile-verified
//
#include <hip/hip_runtime.h>
#include <hip/hip_bf16.h>
#include <math.h>

#define N_NODES 20000
#define N_EDGES 640000
#define DIM 128
#define SOFTMAX_EPS 1e-12f
#define LN_EPS 1e-5f
#define SCALE 0.08838834764831845f  // 128^-0.5

typedef float v2f __attribute__((ext_vector_type(2)));
typedef float v8f __attribute__((ext_vector_type(8)));

// ---------------------------------------------------------------------------
// Core 16x128 @ 128x128 GEMM tile using V_WMMA_F32_16X16X4_F32.
// A: row-major [.,128] fp32, rows [row_base, row_base+16).
// Wsw: weight pre-swizzled into fragment-major float2:
//      Wsw2[(kt*8+nt)*32 + lane] = { W[kt*4 + 2*(lane>>4)    ][nt*16 + (lane&15)],
//                                    W[kt*4 + 2*(lane>>4) + 1][nt*16 + (lane&15)] }
// Produces acc[nt][j] = C[M = j + 8*(lane>>4)][N = nt*16 + (lane&15)]
// per the ISA 16x16 f32 C/D layout.
// ---------------------------------------------------------------------------
__device__ __forceinline__ void gemm_tile_16x128(const float* __restrict__ A,
                                                 long row_base,
                                                 const float* __restrict__ Wsw,
                                                 v8f acc[8]) {
  const int lane = threadIdx.x & 31;
  const int m = lane & 15;
  const int koff = (lane >> 4) << 1;  // 0 or 2
  const float* __restrict__ arow = A + (row_base + m) * DIM;
  const v2f* __restrict__ wfrag = (const v2f*)Wsw;
  for (int kt = 0; kt < 32; ++kt) {
    // A fragment: lane holds A[m][kt*4+koff], A[m][kt*4+koff+1] (8B aligned)
    v2f a = *(const v2f*)(arow + (kt << 2) + koff);
    const v2f* __restrict__ wb = wfrag + (kt << 3) * 32 + lane;
#pragma unroll
    for (int nt = 0; nt < 8; ++nt) {
      v2f b = wb[nt * 32];
      acc[nt] = __builtin_amdgcn_wmma_f32_16x16x4_f32(
          /*neg_a=*/false, a, /*neg_b=*/false, b,
          /*c_mod=*/(short)0, acc[nt], /*reuse_a=*/false, /*reuse_b=*/false);
    }
  }
}

__device__ __forceinline__ void atomicMaxFloat(float* addr, float val) {
  int* ai = (int*)addr;
  int old = __float_as_int(*addr);
  while (__int_as_float(old) < val) {
    int assumed = old;
    old = atomicCAS(ai, assumed, __float_as_int(val));
    if (old == assumed) break;
  }
}

// ---------------------------------------------------------------------------
// Kernel S: swizzle the four 128x128 weights ([in,out] layout) into frag-major.
// ---------------------------------------------------------------------------
__global__ void swizzle_weights_kernel(const float* __restrict__ Wq,
                                       const float* __restrict__ Wk,
                                       const float* __restrict__ Wv,
                                       const float* __restrict__ Wo,
                                       float* __restrict__ ws_w) {
  int t = blockIdx.x * blockDim.x + threadIdx.x;
  if (t >= 4 * 8192) return;
  int w = t >> 13;
  int r = t & 8191;
  int kt = r >> 8;
  int nt = (r >> 5) & 7;
  int lane = r & 31;
  int k = (kt << 2) + ((lane >> 4) << 1);
  int n = (nt << 4) + (lane & 15);
  const float* W = (w == 0) ? Wq : (w == 1) ? Wk : (w == 2) ? Wv : Wo;
  float2 val = make_float2(W[k * DIM + n], W[(k + 1) * DIM + n]);
  ((float2*)(ws_w + w * 16384))[((kt << 3) + nt) * 32 + lane] = val;
}

// ---------------------------------------------------------------------------
// Kernel I: init accumulators (agg=0, node_max=-inf, node_sum=0).
// ---------------------------------------------------------------------------
__global__ void init_kernel(float* __restrict__ node_max,
                            float* __restrict__ node_sum,
                            float* __restrict__ agg) {
  int i = blockIdx.x * blockDim.x + threadIdx.x;
  if (i < N_NODES * DIM) agg[i] = 0.0f;
  if (i < N_NODES) {
    node_max[i] = -INFINITY;
    node_sum[i] = 0.0f;
  }
}

// ---------------------------------------------------------------------------
// Kernel A: q = query_node @ Wq   (one wave per 16-node tile)
// ---------------------------------------------------------------------------
__global__ void q_proj_kernel(const float* __restrict__ query_node,
                              const float* __restrict__ Wq_sw,
                              float* __restrict__ q) {
  int wave = threadIdx.x >> 5;
  int lane = threadIdx.x & 31;
  long tile = (long)blockIdx.x * 4 + wave;
  if (tile >= N_NODES / 16) return;
  long nbase = tile * 16;
  v8f acc[8] = {};
  gemm_tile_16x128(query_node, nbase, Wq_sw, acc);
  int hi = lane >> 4, col = lane & 15;
#pragma unroll
  for (int nt = 0; nt < 8; ++nt)
#pragma unroll
    for (int j = 0; j < 8; ++j)
      q[(nbase + j + 8 * hi) * DIM + nt * 16 + col] = acc[nt][j];
}

// ---------------------------------------------------------------------------
// Kernel B: k = key_edge @ Wk ; scores[e] = dot(q[tgt[e]], k[e]) * scale ;
//           atomic max into node_max[tgt].
// ---------------------------------------------------------------------------
__global__ void k_proj_score_kernel(const float* __restrict__ key_edge,
                                    const float* __restrict__ q,
                                    const int* __restrict__ edge_index,
                                    const float* __restrict__ Wk_sw,
                                    float* __restrict__ scores,
                                    float* __restrict__ node_max) {
  __shared__ float lds[4][16 * 130];
  int wave = threadIdx.x >> 5;
  int lane = threadIdx.x & 31;
  long tile = (long)blockIdx.x * 4 + wave;
  long ebase = tile * 16;
  v8f acc[8] = {};
  gemm_tile_16x128(key_edge, ebase, Wk_sw, acc);

  // spill k-tile to LDS in [m][n] order (stride 130 avoids bank conflicts)
  int hi = lane >> 4, col = lane & 15;
  float* L = lds[wave];
#pragma unroll
  for (int nt = 0; nt < 8; ++nt)
#pragma unroll
    for (int j = 0; j < 8; ++j)
      L[(j + 8 * hi) * 130 + nt * 16 + col] = acc[nt][j];

  // scores: 2 lanes per edge, 64 columns each (same-wave LDS is in-order)
  int me = lane >> 1, h = lane & 1;
  int tgt = edge_index[N_EDGES + ebase + me];  // edge_index[1][e]
  const float* __restrict__ qrow = q + (long)tgt * DIM + h * 64;
  const float* __restrict__ krow = L + me * 130 + h * 64;
  float s = 0.0f;
#pragma unroll 8
  for (int i = 0; i < 64; ++i) s += krow[i] * qrow[i];
  s += __shfl_xor(s, 1);
  if (h == 0) {
    float sc = s * SCALE;
    scores[ebase + me] = sc;
    atomicMaxFloat(&node_max[tgt], sc);
  }
}

// ---------------------------------------------------------------------------
// Kernel B2: e_val = exp(score - m[tgt]); node_sum[tgt] += e_val.
// ---------------------------------------------------------------------------
__global__ void softmax_denom_kernel(const float* __restrict__ scores,
                                     const int* __restrict__ edge_index,
                                     const float* __restrict__ node_max,
                                     float* __restrict__ e_val,
                                     float* __restrict__ node_sum) {
  int e = blockIdx.x * blockDim.x + threadIdx.x;
  if (e >= N_EDGES) return;
  int tgt = edge_index[N_EDGES + e];
  float m = node_max[tgt];
  if (m == -INFINITY) m = 0.0f;  // matches reference isneginf handling
  float ev = expf(scores[e] - m);
  e_val[e] = ev;
  atomicAdd(&node_sum[tgt], ev);
}

// ---------------------------------------------------------------------------
// Kernel C: v = value_edge @ Wv ; attn = e_val/(sum+eps) (written to output);
//           agg[tgt] += attn * v  via global float atomics.
// ---------------------------------------------------------------------------
__global__ void v_proj_scatter_kernel(const float* __restrict__ value_edge,
                                      const int* __restrict__ edge_index,
                                      const float* __restrict__ Wv_sw,
                                      const float* __restrict__ e_val,
                                      const float* __restrict__ node_sum,
                                      float* __restrict__ attn_out,
                                      float* __restrict__ agg) {
  __shared__ float lds_attn[4][16];
  __shared__ int lds_tgt[4][16];
  int wave = threadIdx.x >> 5;
  int lane = threadIdx.x & 31;
  long tile = (long)blockIdx.x * 4 + wave;
  long ebase = tile * 16;
  v8f acc[8] = {};
  gemm_tile_16x128(value_edge, ebase, Wv_sw, acc);

  if (lane < 16) {
    long e = ebase + lane;
    int tgt = edge_index[N_EDGES + e];
    float a = e_val[e] / (node_sum[tgt] + SOFTMAX_EPS);
    attn_out[e] = a;
    lds_attn[wave][lane] = a;
    lds_tgt[wave][lane] = tgt;
  }
  // same-wave LDS RAW: DS pipe is in-order within a wave
  int hi = lane >> 4, col = lane & 15;
#pragma unroll
  for (int j = 0; j < 8; ++j) {
    int m = j + 8 * hi;
    float a = lds_attn[wave][m];
    int tgt = lds_tgt[wave][m];
    float* dst = agg + (long)tgt * DIM + col;
#pragma unroll
    for (int nt = 0; nt < 8; ++nt)
      atomicAdd(dst + nt * 16, a * acc[nt][j]);
  }
}

// ---------------------------------------------------------------------------
// Kernel D: proj = agg @ Wo + bo ; LayerNorm(gamma, beta) -> out.
// ---------------------------------------------------------------------------
__global__ void out_proj_ln_kernel(const float* __restrict__ agg,
                                   const float* __restrict__ Wo_sw,
                                   const float* __restrict__ bo,
                                   const float* __restrict__ gamma,
                                   const float* __restrict__ beta,
                                   float* __restrict__ out) {
  __shared__ float lds[4][16 * 130];
  int wave = threadIdx.x >> 5;
  int lane = threadIdx.x & 31;
  long tile = (long)blockIdx.x * 4 + wave;
  if (tile >= N_NODES / 16) return;
  long nbase = tile * 16;
  v8f acc[8] = {};
  gemm_tile_16x128(agg, nbase, Wo_sw, acc);

  int hi = lane >> 4, col = lane & 15;
  float* L = lds[wave];
#pragma unroll
  for (int nt = 0; nt < 8; ++nt)
#pragma unroll
    for (int j = 0; j < 8; ++j)
      L[(j + 8 * hi) * 130 + nt * 16 + col] = acc[nt][j];

  // LayerNorm: 2 lanes per row
  int m = lane >> 1, h = lane & 1;
  const float* __restrict__ row = L + m * 130 + h * 64;
  float sum = 0.0f, sumsq = 0.0f;
#pragma unroll 8
  for (int i = 0; i < 64; ++i) {
    float x = row[i] + bo[h * 64 + i];
    sum += x;
    sumsq += x * x;
  }
  sum += __shfl_xor(sum, 1);
  sumsq += __shfl_xor(sumsq, 1);
  float mu = sum * (1.0f / 128.0f);
  float var = sumsq * (1.0f / 128.0f) - mu * mu;
  float inv = rsqrtf(var + LN_EPS);
  float* __restrict__ orow = out + (nbase + m) * DIM + h * 64;
#pragma unroll 8
  for (int i = 0; i < 64; ++i) {
    int c = h * 64 + i;
    float x = row[i] + bo[c];
    orow[i] = (x - mu) * inv * gamma[c] + beta[c];
  }
}

// ---------------------------------------------------------------------------
// Launch
// ---------------------------------------------------------------------------
extern "C" void kernel_launch(void* const* d_in, const int* in_sizes, int n_in,
                              void* d_out, int out_size, void* d_ws, size_t ws_size,
                              hipStream_t stream) {
  const float* query_node = (const float*)d_in[0];
  const float* key_edge   = (const float*)d_in[1];
  const float* value_edge = (const float*)d_in[2];
  const int*   edge_index = (const int*)d_in[3];
  const float* Wq = (const float*)d_in[4];
  const float* Wk = (const float*)d_in[5];
  const float* Wv = (const float*)d_in[6];
  const float* Wo = (const float*)d_in[7];
  const float* bo = (const float*)d_in[8];
  const float* gamma = (const float*)d_in[9];
  const float* beta  = (const float*)d_in[10];

  float* out_ptr  = (float*)d_out;                      // [N_NODES*128]
  float* attn_ptr = out_ptr + (size_t)N_NODES * DIM;    // [N_EDGES]

  // workspace layout (floats)
  float* w = (float*)d_ws;
  float* Wq_sw = w;                 // 16384
  float* Wk_sw = Wq_sw + 16384;
  float* Wv_sw = Wk_sw + 16384;
  float* Wo_sw = Wv_sw + 16384;
  float* q_buf = Wo_sw + 16384;                       // N_NODES*128
  float* agg   = q_buf + (size_t)N_NODES * DIM;       // N_NODES*128
  float* scores = agg + (size_t)N_NODES * DIM;        // N_EDGES
  float* e_val  = scores + N_EDGES;                   // N_EDGES
  float* node_max = e_val + N_EDGES;                  // N_NODES
  float* node_sum = node_max + N_NODES;               // N_NODES

  const int E_TILES = N_EDGES / 16;   // 40000
  const int N_TILES = N_NODES / 16;   // 1250

  swizzle_weights_kernel<<<(4 * 8192 + 255) / 256, 256, 0, stream>>>(
      Wq, Wk, Wv, Wo, Wq_sw);
  init_kernel<<<(N_NODES * DIM + 255) / 256, 256, 0, stream>>>(
      node_max, node_sum, agg);
  q_proj_kernel<<<(N_TILES + 3) / 4, 128, 0, stream>>>(query_node, Wq_sw, q_buf);
  k_proj_score_kernel<<<E_TILES / 4, 128, 0, stream>>>(
      key_edge, q_buf, edge_index, Wk_sw, scores, node_max);
  softmax_denom_kernel<<<(N_EDGES + 255) / 256, 256, 0, stream>>>(
      scores, edge_index, node_max, e_val, node_sum);
  v_proj_scatter_kernel<<<E_TILES / 4, 128, 0, stream>>>(
      value_edge, edge_index, Wv_sw, e_val, node_sum, attn_ptr, agg);
  out_proj_ln_kernel<<<(N_TILES + 3) / 4, 128, 0, stream>>>(
      agg, Wo_sw, bo, gamma, beta, out_ptr);
}